// SDFCLayer_65773129171730
// MI455X (gfx1250) — compile-verified
//
#include <hip/hip_runtime.h>

// CDNA5 (gfx1250) wave32 WMMA edge-MLP:
//   out[e] = relu(z[src[e]] @ W1[:H] + z[dst[e]] @ W1[H:] + b1) @ W2 + b2
// Exact-f32 path via V_WMMA_F32_16X16X4_F32; W1 staged in LDS in WMMA
// B-fragment order for conflict-free ds_load_b64.

typedef __attribute__((ext_vector_type(2))) float v2f;
typedef __attribute__((ext_vector_type(8))) float v8f;

#define HDIM   128
#define TILE_M 16
#define MB     2            // edge-tiles per wave iteration (32 edges)
#define NT     8            // 128 / 16 column tiles
#define KSTEPS 32           // 128 / 4
#define NFRAG  (2 * NT * KSTEPS)        // src-block + dst-block fragments = 512
#define LDS_V2F (NFRAG * 32)            // 16384 v2f = 128 KB

// fragment offset (in v2f elements) for (mat, nt, t)
#define FOFF(mat, nt, t) ((((mat) * NT + (nt)) * KSTEPS + (t)) * 32)

__global__ __launch_bounds__(256)
void edge_mlp_wmma_f32(const float* __restrict__ z,
                       const long long* __restrict__ eidx,   // [2, E] int64
                       const float* __restrict__ W1,         // [256, 128] row-major
                       const float* __restrict__ b1,         // [128]
                       const float* __restrict__ W2,         // [128]
                       const float* __restrict__ b2,         // [1]
                       float* __restrict__ out,              // [E]
                       int E, int npairs)
{
    extern __shared__ v2f frag[];   // 128 KB: W1 in WMMA B-fragment layout

    const int tid = threadIdx.x;

    // ---- Stage W1 into LDS, pre-swizzled to B-fragment order ----
    // element i <-> (mat, nt, t, lane); lane L<16: N=L, K=t*4+{0,1};
    //                                   lane L>=16: N=L-16, K=t*4+{2,3}
    for (int i = tid; i < LDS_V2F; i += blockDim.x) {
        const int lane = i & 31;
        const int t    = (i >> 5) & (KSTEPS - 1);
        const int nt   = (i >> 10) & (NT - 1);
        const int mat  = (i >> 13) & 1;
        const int half = lane >> 4;
        const int n    = nt * 16 + (lane & 15);
        const int k    = mat * HDIM + t * 4 + 2 * half;
        v2f v;
        v.x = W1[k * HDIM + n];
        v.y = W1[(k + 1) * HDIM + n];
        frag[i] = v;
    }
    __syncthreads();

    const int lane = tid & 31;
    const int half = lane >> 4;     // 0: K lanes 0/1, rows M=0..7 of C
    const int r    = lane & 15;     // A row / C column within tile

    // per-lane epilogue constants: columns n = nt*16 + r
    float rb1[NT], rw2[NT];
#pragma unroll
    for (int nt = 0; nt < NT; ++nt) {
        const int col = nt * 16 + r;
        rb1[nt] = b1[col];
        rw2[nt] = W2[col];
    }
    const float b2v = b2[0];

    const int wavesPerBlock = blockDim.x >> 5;
    const int wid    = blockIdx.x * wavesPerBlock + (tid >> 5);
    const int nwaves = gridDim.x * wavesPerBlock;

    for (int pair = wid; pair < npairs; pair += nwaves) {
        const int eBase = pair * (MB * TILE_M);

        const v2f* srow[MB];
        const v2f* drow[MB];
#pragma unroll
        for (int mb = 0; mb < MB; ++mb) {
            const int e = eBase + mb * TILE_M + r;
            const long long si = (e < E) ? eidx[e] : 0ll;
            const long long di = (e < E) ? eidx[(long long)E + e] : 0ll;
            srow[mb] = (const v2f*)(z + si * HDIM);
            drow[mb] = (const v2f*)(z + di * HDIM);
        }

        v8f acc[MB][NT];
#pragma unroll
        for (int mb = 0; mb < MB; ++mb)
#pragma unroll
            for (int nt = 0; nt < NT; ++nt)
                acc[mb][nt] = (v8f){};

        for (int t = 0; t < KSTEPS; ++t) {
            v2f as[MB], ad[MB];
#pragma unroll
            for (int mb = 0; mb < MB; ++mb) {
                as[mb] = srow[mb][t * 2 + half];  // K = t*4 + 2*half + {0,1}
                ad[mb] = drow[mb][t * 2 + half];
            }
#pragma unroll
            for (int nt = 0; nt < NT; ++nt) {
                const v2f bs = frag[FOFF(0, nt, t) + lane];  // W1[:H] block
                const v2f bd = frag[FOFF(1, nt, t) + lane];  // W1[H:] block
#pragma unroll
                for (int mb = 0; mb < MB; ++mb) {
                    acc[mb][nt] = __builtin_amdgcn_wmma_f32_16x16x4_f32(
                        false, as[mb], false, bs, (short)0, acc[mb][nt], false, false);
                    acc[mb][nt] = __builtin_amdgcn_wmma_f32_16x16x4_f32(
                        false, ad[mb], false, bd, (short)0, acc[mb][nt], false, false);
                }
            }
        }

        // ---- epilogue: bias + relu + dot(W2), reduce across 16-lane halves ----
#pragma unroll
        for (int mb = 0; mb < MB; ++mb) {
            float p[8];
#pragma unroll
            for (int g = 0; g < 8; ++g) p[g] = 0.0f;
#pragma unroll
            for (int nt = 0; nt < NT; ++nt) {
#pragma unroll
                for (int g = 0; g < 8; ++g) {
                    float h = acc[mb][nt][g] + rb1[nt];   // C: M = 8*half+g, N = nt*16+r
                    h = h > 0.0f ? h : 0.0f;
                    p[g] += h * rw2[nt];
                }
            }
#pragma unroll
            for (int mask = 1; mask < 16; mask <<= 1) {
#pragma unroll
                for (int g = 0; g < 8; ++g)
                    p[g] += __shfl_xor(p[g], mask, 32);
            }
            if (r == 0) {
                const int e0 = eBase + mb * TILE_M + half * 8;
#pragma unroll
                for (int g = 0; g < 8; ++g) {
                    const int e2 = e0 + g;
                    if (e2 < E) out[e2] = p[g] + b2v;
                }
            }
        }
    }
}

extern "C" void kernel_launch(void* const* d_in, const int* in_sizes, int n_in,
                              void* d_out, int out_size, void* d_ws, size_t ws_size,
                              hipStream_t stream) {
    const float*     z    = (const float*)d_in[0];
    const long long* eidx = (const long long*)d_in[1];   // int64 [2, E]
    const float*     W1   = (const float*)d_in[2];
    const float*     b1   = (const float*)d_in[3];
    const float*     W2   = (const float*)d_in[4];
    const float*     b2   = (const float*)d_in[5];
    float*           out  = (float*)d_out;

    const int E      = in_sizes[1] / 2;
    const int ntiles = (E + TILE_M - 1) / TILE_M;
    const int npairs = (ntiles + MB - 1) / MB;

    const int block  = 256;                    // 8 wave32s sharing one LDS W1 copy
    const int blocks = 1024;                   // persistent-style grid
    const size_t ldsBytes = LDS_V2F * sizeof(v2f);   // 131072 B (<= 320KB/WG on CDNA5)

    edge_mlp_wmma_f32<<<blocks, block, ldsBytes, stream>>>(
        z, eidx, W1, b1, W2, b2, out, E, npairs);
}